// JointLayer_66340064854456
// MI455X (gfx1250) — compile-verified
//
#include <hip/hip_runtime.h>

typedef __attribute__((ext_vector_type(2))) float v2f;
typedef __attribute__((ext_vector_type(8))) float v8f;

// Problem constants from the reference
#define N_LINKS   33
#define N_JOINTS  32
#define KDIM      1024          // C_L * 16
#define NDIM      128           // C_J
#define KT        16            // K tile staged in LDS per stage
#define A_STRIDE  20            // dwords per A row (16 + 4 pad): 16B aligned, conflict-free
#define B_STRIDE  288           // dwords per B k-pair row = 2*(128+16): +32 banks per row

// ---- gfx1250 async global->LDS DMA (ASYNCcnt-tracked) ----
__device__ __forceinline__ void async_copy_b128(unsigned lds_addr, const float* gaddr) {
    asm volatile("global_load_async_to_lds_b128 %0, %1, off"
                 :: "v"(lds_addr), "v"(gaddr) : "memory");
}
__device__ __forceinline__ void async_copy_b32(unsigned lds_addr, const float* gaddr) {
    asm volatile("global_load_async_to_lds_b32 %0, %1, off"
                 :: "v"(lds_addr), "v"(gaddr) : "memory");
}
__device__ __forceinline__ void wait_async0() {
    asm volatile("s_wait_asynccnt 0x0" ::: "memory");
}

__global__ __launch_bounds__(256, 2)
void joint_grouped_gemm_wmma(const float* __restrict__ link_feats,
                             const float* __restrict__ joint_feats,
                             const float* __restrict__ W,
                             const float* __restrict__ bias,
                             const int*   __restrict__ child_raw,
                             float* __restrict__ out)
{
    // A: [buf][m][k]           (stride-20 rows)
    // B: [buf][k/2][col*2 + e] (k-pair interleaved so a B fragment is one b64)
    __shared__ float As[2][128][A_STRIDE];
    __shared__ float Bs[2][KT / 2][B_STRIDE];

    const int tid = threadIdx.x;
    const int j   = blockIdx.y;            // joint
    const int m0  = blockIdx.x * 128;      // batch-row tile base

    // child_idx may arrive as int64 (reference dtype) or canonicalized int32.
    const bool is64 = (child_raw[1] == 0);
    const int child = is64 ? child_raw[2 * j] : child_raw[j];

    const float* gA = link_feats + ((size_t)m0 * N_LINKS + child) * KDIM;
    const float* gB = W + (size_t)j * (KDIM * NDIM);

    const int lane  = tid & 31;
    const int wv    = tid >> 5;
    const int mg    = wv >> 1;             // 0..3 : 32-row group
    const int ng    = wv & 1;              // 0..1 : 64-col group
    const int lhalf = (lane >> 4) << 1;    // 0 or 2 (K offset per WMMA lane layout)
    const int l15   = lane & 15;

    const unsigned aBase = (unsigned)(size_t)&As[0][0][0];
    const unsigned bBase = (unsigned)(size_t)&Bs[0][0][0];
    const unsigned aBufBytes = 128 * A_STRIDE * 4;
    const unsigned bBufBytes = (KT / 2) * B_STRIDE * 4;

    // ---- issue one K-stage of async DMA into LDS buffer `buf` ----
    auto issue_stage = [&](int kt, int buf) {
        const unsigned aB = aBase + (unsigned)buf * aBufBytes;
        const unsigned bB = bBase + (unsigned)buf * bBufBytes;
        // A tile: 128 rows x 16 floats -> 2 x b128 per thread
#pragma unroll
        for (int i = 0; i < 2; ++i) {
            int q   = tid + i * 256;
            int row = q >> 2, c4 = (q & 3) << 2;
            unsigned lds = aB + (((unsigned)(row * A_STRIDE + c4)) << 2);
            async_copy_b128(lds, gA + (size_t)row * (N_LINKS * KDIM) + kt + c4);
        }
        // B tile: 16 k-rows x 128 cols, written k-pair interleaved -> 8 x b32 per thread
#pragma unroll
        for (int i = 0; i < 4; ++i) {
            int p   = tid + i * 256;
            int ks2 = p >> 7, col = p & 127;
            unsigned lds0 = bB + (((unsigned)(ks2 * B_STRIDE + col * 2)) << 2);
            const float* g0 = gB + (size_t)(kt + ks2 * 2) * NDIM + col;
            async_copy_b32(lds0,     g0);         // k = kt + 2*ks2
            async_copy_b32(lds0 + 4, g0 + NDIM);  // k = kt + 2*ks2 + 1
        }
    };

    v8f acc[2][4];
#pragma unroll
    for (int mt = 0; mt < 2; ++mt)
#pragma unroll
        for (int nt = 0; nt < 4; ++nt)
            acc[mt][nt] = (v8f){0.f, 0.f, 0.f, 0.f, 0.f, 0.f, 0.f, 0.f};

    issue_stage(0, 0);
    int buf = 0;

    for (int kt = 0; kt < KDIM; kt += KT) {
        wait_async0();        // this wave's DMA into LDS buffer buf has landed
        __syncthreads();      // all waves' DMA landed; prior-stage compute done
        if (kt + KT < KDIM)
            issue_stage(kt + KT, buf ^ 1);   // DMA next stage, overlapped with WMMAs

        // ---- compute: 4 k-steps x (2x4) WMMA tiles = 32 WMMA per wave ----
#pragma unroll
        for (int ks = 0; ks < KT; ks += 4) {
            const int kb = ks + lhalf;       // lanes 0-15: K=ks,ks+1 ; lanes 16-31: +2
            v2f afr[2];
#pragma unroll
            for (int mt = 0; mt < 2; ++mt)
                afr[mt] = *reinterpret_cast<const v2f*>(
                    &As[buf][mg * 32 + mt * 16 + l15][kb]);
            v2f bfr[4];
            const int kp = kb >> 1;
#pragma unroll
            for (int nt = 0; nt < 4; ++nt)
                bfr[nt] = *reinterpret_cast<const v2f*>(
                    &Bs[buf][kp][(ng * 64 + nt * 16 + l15) * 2]);
#pragma unroll
            for (int mt = 0; mt < 2; ++mt)
#pragma unroll
                for (int nt = 0; nt < 4; ++nt)
                    acc[mt][nt] = __builtin_amdgcn_wmma_f32_16x16x4_f32(
                        false, afr[mt], false, bfr[nt],
                        (short)0, acc[mt][nt], false, false);
        }
        buf ^= 1;
    }

    // ---- epilogue: + bias + residual (joint_feats), fused store ----
#pragma unroll
    for (int mt = 0; mt < 2; ++mt) {
        const int rbase = m0 + mg * 32 + mt * 16 + ((lane >> 4) << 3);
#pragma unroll
        for (int nt = 0; nt < 4; ++nt) {
            const int o  = ng * 64 + nt * 16 + l15;
            const float bv = bias[(size_t)j * NDIM + o];
#pragma unroll
            for (int v = 0; v < 8; ++v) {
                const size_t idx = (size_t)(rbase + v) * (N_JOINTS * NDIM)
                                 + (size_t)j * NDIM + o;
                out[idx] = acc[mt][nt][v] + bv + joint_feats[idx];
            }
        }
    }
}

extern "C" void kernel_launch(void* const* d_in, const int* in_sizes, int n_in,
                              void* d_out, int out_size, void* d_ws, size_t ws_size,
                              hipStream_t stream) {
    const float* link_feats  = (const float*)d_in[0];
    const float* joint_feats = (const float*)d_in[1];
    const float* W           = (const float*)d_in[2];
    const float* bias        = (const float*)d_in[3];
    const int*   child_raw   = (const int*)d_in[4];
    float* out = (float*)d_out;

    const int Bsz = in_sizes[0] / (N_LINKS * KDIM);   // link_feats: [B,33,64,16]
    dim3 grid(Bsz / 128, N_JOINTS);
    joint_grouped_gemm_wmma<<<grid, 256, 0, stream>>>(
        link_feats, joint_feats, W, bias, child_raw, out);
}